// LearnableHistEq_81355270521054
// MI455X (gfx1250) — compile-verified
//
#include <hip/hip_runtime.h>
#include <math.h>

#define NBINS   64
#define GROUPS  16
#define HIDDEN  128
#define EPSF    1e-6f

typedef __attribute__((ext_vector_type(2))) float v2f;
typedef __attribute__((ext_vector_type(4))) float v4f;
typedef __attribute__((ext_vector_type(8))) float v8f;

// Workspace layout (float offsets)
#define WS_BLOCKSUM   0        // 32 * 1024   per-(b,c) 32x32 block sums of raw x
#define WS_STRIPMIN   32768    // 32 * 32     per-(b,c,striprow) min
#define WS_STRIPMAX   33792    // 32 * 32
#define WS_CHANMIN    34816    // 32
#define WS_CHANMAX    34848    // 32
#define WS_DCDF       34880    // 16 * 64
#define WS_LUT        35904    // 16 * 64

// ---------------------------------------------------------------------------
// Kernel 1: per 32-row strip of one (b,c) channel: strip min/max + 32 block sums.
// 1024 blocks (32 channels * 32 strips), 256 threads. Streams x once (fills L2).
// ---------------------------------------------------------------------------
__global__ void k1_reduce(const float* __restrict__ x, float* __restrict__ ws) {
  const int bc    = blockIdx.x >> 5;   // 0..31  (b*16 + c)
  const int strip = blockIdx.x & 31;   // 0..31  (block-row of 32 rows)
  const int t   = threadIdx.x;
  const int blk = t >> 3;              // which 32-col block (0..31)
  const int sub = t & 7;               // which float4 inside the block's row
  const v4f* x4 = (const v4f*)x;
  const int base = bc * 262144 + strip * 32 * 256 + blk * 8 + sub;

  float s = 0.0f, mn = __FLT_MAX__, mx = -__FLT_MAX__;
#pragma unroll 4
  for (int r = 0; r < 32; ++r) {
    v4f v = x4[base + r * 256];
    s  += (v.x + v.y) + (v.z + v.w);
    mn  = fminf(mn, fminf(fminf(v.x, v.y), fminf(v.z, v.w)));
    mx  = fmaxf(mx, fmaxf(fmaxf(v.x, v.y), fmaxf(v.z, v.w)));
  }

  __shared__ float psum[256], pmin[256], pmax[256];
  psum[t] = s; pmin[t] = mn; pmax[t] = mx;
  __syncthreads();

  if (t < 32) {                       // 8 partials -> one 32x32 block sum
    float bs = 0.0f;
    for (int j = 0; j < 8; ++j) bs += psum[t * 8 + j];
    ws[WS_BLOCKSUM + bc * 1024 + strip * 32 + t] = bs;
  }
  __syncthreads();
  for (int sft = 128; sft > 0; sft >>= 1) {
    if (t < sft) {
      pmin[t] = fminf(pmin[t], pmin[t + sft]);
      pmax[t] = fmaxf(pmax[t], pmax[t + sft]);
    }
    __syncthreads();
  }
  if (t == 0) {
    ws[WS_STRIPMIN + bc * 32 + strip] = pmin[0];
    ws[WS_STRIPMAX + bc * 32 + strip] = pmax[0];
  }
}

// ---------------------------------------------------------------------------
// Kernel 2: channel min/max, downsampled x01, histogram, batch-mean CDF.
// Single block, 256 threads.
// ---------------------------------------------------------------------------
__global__ void k2_stats(float* __restrict__ ws) {
  const int t = threadIdx.x;
  __shared__ float cmin[32], cmax[32];
  __shared__ int   hist[2 * GROUPS * NBINS];   // [b*16+g][64]

  if (t < 32) {
    float mn = __FLT_MAX__, mx = -__FLT_MAX__;
    for (int s = 0; s < 32; ++s) {
      mn = fminf(mn, ws[WS_STRIPMIN + t * 32 + s]);
      mx = fmaxf(mx, ws[WS_STRIPMAX + t * 32 + s]);
    }
    cmin[t] = mn; cmax[t] = mx;
    ws[WS_CHANMIN + t] = mn;
    ws[WS_CHANMAX + t] = mx;
  }
  for (int i = t; i < 2 * GROUPS * NBINS; i += 256) hist[i] = 0;
  __syncthreads();

  // block-mean(x01) = (block-mean(x) - min) / (max - min + eps); C/G == 1 so g == c
  for (int i = t; i < 32768; i += 256) {
    const int bc = i >> 10;
    const float m  = ws[WS_BLOCKSUM + i] * (1.0f / 1024.0f);
    const float xs = (m - cmin[bc]) / (cmax[bc] - cmin[bc] + EPSF);
    int idx = (int)rintf(xs * 63.0f);          // RNE matches jnp.round
    idx = max(0, min(63, idx));
    atomicAdd(&hist[bc * NBINS + idx], 1);
  }
  __syncthreads();

  if (t < GROUPS) {                            // cdf per (b,g), mean over b
    float a0 = 0.0f, a1 = 0.0f;
    const float inv = 1.0f / (1024.0f + EPSF); // hist.sum == 1024 exactly
    for (int k = 0; k < NBINS; ++k) {
      a0 += (float)hist[t * NBINS + k];
      a1 += (float)hist[(GROUPS + t) * NBINS + k];
      ws[WS_DCDF + t * NBINS + k] = 0.5f * (a0 + a1) * inv;
    }
  }
}

// ---------------------------------------------------------------------------
// Kernel 3: LUT net. conv1 & conv3 as f32 WMMA GEMMs, conv2 depthwise VALU.
// Single block, 256 threads = 8 waves.
//   conv1: out(128x64) = W1eff(128x80) * dcdf_pad(80x64)   (K = ic*5 + kw)
//   conv3: out(16x64)  = W3(16x128)    * h2(128x64)
// ---------------------------------------------------------------------------
__global__ void k3_lutnet(const float* __restrict__ w1, const float* __restrict__ b1,
                          const float* __restrict__ w2, const float* __restrict__ b2,
                          const float* __restrict__ w3, const float* __restrict__ b3,
                          float* __restrict__ ws) {
  __shared__ float sH1[HIDDEN * 64];  // conv1 output (post-ReLU)
  __shared__ float sB [HIDDEN * 64];  // dcdf (first 1024 floats), then conv2 output

  const int t    = threadIdx.x;
  const int wave = t >> 5;
  const int lane = t & 31;
  const int lm   = lane & 15;         // m / n within a 16-wide tile
  const int lh   = lane >> 4;         // half-wave selector

  for (int i = t; i < GROUPS * NBINS; i += 256) sB[i] = ws[WS_DCDF + i];
  __syncthreads();

  // ---- conv1 via V_WMMA_F32_16X16X4_F32: wave w owns out rows [16w,16w+16)
  {
    v8f acc[4] = {};
    const int oc = wave * 16 + lm;    // A-matrix row for this lane
    for (int kt = 0; kt < 20; ++kt) { // K = 80 = 20 * 4
      const int k0  = kt * 4 + lh * 2;       // ISA 16x4 f32 A/B layout
      const int k1  = k0 + 1;
      const int ic0 = k0 / 5, kw0 = k0 % 5;
      const int ic1 = k1 / 5, kw1 = k1 % 5;
      v2f a;
      a.x = w1[oc * 400 + ic0 * 25 + 10 + kw0];  // center kernel row (kh==2)
      a.y = w1[oc * 400 + ic1 * 25 + 10 + kw1];
#pragma unroll
      for (int nt = 0; nt < 4; ++nt) {
        const int c0 = nt * 16 + lm + kw0 - 2;   // zero-padded bin axis
        const int c1 = nt * 16 + lm + kw1 - 2;
        v2f b;
        b.x = (c0 >= 0 && c0 < 64) ? sB[ic0 * 64 + c0] : 0.0f;
        b.y = (c1 >= 0 && c1 < 64) ? sB[ic1 * 64 + c1] : 0.0f;
        acc[nt] = __builtin_amdgcn_wmma_f32_16x16x4_f32(
            false, a, false, b, (short)0, acc[nt], false, false);
      }
    }
#pragma unroll
    for (int nt = 0; nt < 4; ++nt) {
#pragma unroll
      for (int j = 0; j < 8; ++j) {
        const int m = wave * 16 + lh * 8 + j;   // C/D layout: m = j + 8*half
        const int n = nt * 16 + lm;
        sH1[m * 64 + n] = fmaxf(acc[nt][j] + b1[m], 0.0f);
      }
    }
  }
  __syncthreads();

  // ---- conv2: depthwise 5-tap + ReLU  (overwrites dcdf region; dcdf is dead)
  for (int i = t; i < HIDDEN * 64; i += 256) {
    const int ch = i >> 6, w = i & 63;
    float s = b2[ch];
#pragma unroll
    for (int kw = 0; kw < 5; ++kw) {
      const int c = w + kw - 2;
      if (c >= 0 && c < 64) s += w2[ch * 25 + 10 + kw] * sH1[ch * 64 + c];
    }
    sB[i] = fmaxf(s, 0.0f);
  }
  __syncthreads();

  // ---- conv3 via WMMA: waves 0..3 each own one 16-col N tile (EXEC full per wave)
  if (wave < 4) {
    const int nt = wave;
    v8f acc = {};
    for (int kt = 0; kt < 32; ++kt) {  // K = 128 = 32 * 4
      const int k0 = kt * 4 + lh * 2;
      v2f a, b;
      a.x = w3[lm * 128 + k0];
      a.y = w3[lm * 128 + k0 + 1];
      b.x = sB[k0 * 64 + nt * 16 + lm];
      b.y = sB[(k0 + 1) * 64 + nt * 16 + lm];
      acc = __builtin_amdgcn_wmma_f32_16x16x4_f32(
          false, a, false, b, (short)0, acc, false, false);
    }
#pragma unroll
    for (int j = 0; j < 8; ++j) {
      const int g = lh * 8 + j;
      const int n = nt * 16 + lm;
      const float d  = acc[j] + b3[g];
      const float sp = fmaxf(d, 0.0f) + log1pf(expf(-fabsf(d)));  // softplus
      sH1[g * 64 + n] = sp;
    }
  }
  __syncthreads();

  // ---- cumsum, normalize, + identity -> LUT (16 x 64)
  if (t < GROUPS) {
    float accu = 0.0f;
    for (int k = 0; k < NBINS; ++k) { accu += sH1[t * 64 + k]; sH1[t * 64 + k] = accu; }
    const float inv = 1.0f / (accu + EPSF);
    for (int k = 0; k < NBINS; ++k)
      ws[WS_LUT + t * NBINS + k] = sH1[t * 64 + k] * inv + (float)k * (1.0f / 63.0f);
  }
}

// ---------------------------------------------------------------------------
// Kernel 4: streaming apply. x read is expected to hit L2 (filled by k1);
// output written with non-temporal stores so it doesn't evict x.
// grid (64, 32): y = (b,c) channel, x = 16384-pixel chunk. 256 threads.
// ---------------------------------------------------------------------------
__device__ __forceinline__ float apply_one(float xv, float mn, float invr, float rng,
                                           float a, float oma,
                                           const float* __restrict__ slut) {
  const float x01 = (xv - mn) * invr;
  const float pos = x01 * 63.0f;
  int il = (int)floorf(pos);
  il = max(0, min(63, il));
  const int ih = min(il + 1, 63);
  const float wh = pos - (float)il;
  const float vl = slut[il];
  const float vh = slut[ih];
  const float xe = vl + wh * (vh - vl);
  return (a * xe + oma * x01) * rng + mn;
}

__global__ void k4_apply(const float* __restrict__ x, const float* __restrict__ ws,
                         const float* __restrict__ alpha_p, float* __restrict__ out) {
  const int bc = blockIdx.y;          // b*16 + c
  const int c  = bc & 15;
  const int t  = threadIdx.x;
  __shared__ float slut[NBINS];
  if (t < NBINS) slut[t] = ws[WS_LUT + c * NBINS + t];
  __syncthreads();

  const float mn   = ws[WS_CHANMIN + bc];
  const float mx   = ws[WS_CHANMAX + bc];
  const float rng  = mx - mn;                    // reference rescales without eps
  const float invr = 1.0f / (rng + EPSF);        // but normalizes with eps
  const float a    = 1.0f / (1.0f + expf(-alpha_p[0]));
  const float oma  = 1.0f - a;

  const v4f* x4 = (const v4f*)x;
  v4f*       o4 = (v4f*)out;
  const int base = bc * 262144 + blockIdx.x * 4096;

#pragma unroll 4
  for (int it = 0; it < 16; ++it) {
    const int i = base + it * 256 + t;
    const v4f v = x4[i];
    v4f o;
    o.x = apply_one(v.x, mn, invr, rng, a, oma, slut);
    o.y = apply_one(v.y, mn, invr, rng, a, oma, slut);
    o.z = apply_one(v.z, mn, invr, rng, a, oma, slut);
    o.w = apply_one(v.w, mn, invr, rng, a, oma, slut);
    __builtin_nontemporal_store(o, &o4[i]);
  }
}

// ---------------------------------------------------------------------------
extern "C" void kernel_launch(void* const* d_in, const int* in_sizes, int n_in,
                              void* d_out, int out_size, void* d_ws, size_t ws_size,
                              hipStream_t stream) {
  const float* x     = (const float*)d_in[0];
  const float* w1    = (const float*)d_in[1];
  const float* b1    = (const float*)d_in[2];
  const float* w2    = (const float*)d_in[3];
  const float* b2    = (const float*)d_in[4];
  const float* w3    = (const float*)d_in[5];
  const float* b3    = (const float*)d_in[6];
  const float* alpha = (const float*)d_in[7];
  float* ws  = (float*)d_ws;
  float* out = (float*)d_out;

  hipLaunchKernelGGL(k1_reduce, dim3(1024), dim3(256), 0, stream, x, ws);
  hipLaunchKernelGGL(k2_stats,  dim3(1),    dim3(256), 0, stream, ws);
  hipLaunchKernelGGL(k3_lutnet, dim3(1),    dim3(256), 0, stream,
                     w1, b1, w2, b2, w3, b3, ws);
  hipLaunchKernelGGL(k4_apply,  dim3(64, 32), dim3(256), 0, stream, x, ws, alpha, out);
}